// SinusoidalPositionalEmbedding_15290083573994
// MI455X (gfx1250) — compile-verified
//
#include <hip/hip_runtime.h>
#include <stdint.h>

// SinusoidalPositionalEmbedding concat, async-copy-engine version:
//   out[b,s,0:512] = inp[b,s,:]
//   out[b,s,512]   = (s < len[b]) ? cosf((s / max(len[b],1)) * pi) : 0
//
// Streaming op, ~269 MB traffic -> ~11.5 us floor at 23.3 TB/s. The 512->513
// row-stride conversion is done via LDS staging with the gfx1250 async
// global<->LDS copy path (ASYNCcnt): b128 async loads fill padded LDS rows,
// a ds_store injects the positional scalar, b32 async stores drain the
// contiguous 513-float output rows. No matrix math => no WMMA by design.

#define PE_PI_F 3.14159265358979323846f

#define ROWS_PER_BLOCK 16u
#define LDS_ROW_F      516u   // padded row stride in floats (2064 B, 16B-aligned)

__global__ __launch_bounds__(256) void concat_pe_async_kernel(
    const float* __restrict__ inp,      // [65536 x 512]
    const int*   __restrict__ lengths,  // [16]
    float*       __restrict__ out)      // [65536 x 513]
{
    __shared__ __align__(16) float tile[ROWS_PER_BLOCK * LDS_ROW_F];

    const uint32_t t        = threadIdx.x;
    const uint32_t row0     = blockIdx.x * ROWS_PER_BLOCK;
    const uint32_t lds_base = (uint32_t)(uintptr_t)tile;  // low 32b of generic LDS ptr = LDS offset

    // ---- Phase 1: async global -> LDS, 16B per lane per issue -------------
    // Block input = 16 rows * 2048 B = 2048 chunks of 16 B; 8 issues/thread.
#pragma unroll
    for (uint32_t j = 0; j < 8; ++j) {
        uint32_t c   = t + j * 256u;                 // chunk id 0..2047
        uint32_t r   = c >> 7;                       // 128 chunks per row
        uint32_t k   = c & 127u;
        uint32_t lds = lds_base + r * (LDS_ROW_F * 4u) + k * 16u;
        uint32_t gof = (row0 + r) * 2048u + k * 16u; // byte offset into inp
        asm volatile("global_load_async_to_lds_b128 %0, %1, %2 th:TH_LOAD_NT"
                     :: "v"(lds), "v"(gof), "s"(inp) : "memory");
    }

    // ---- Positional scalar into LDS slot d=512 of each row ----------------
    if (t < ROWS_PER_BLOCK) {
        uint32_t grow = row0 + t;
        uint32_t b    = grow >> 12;                  // batch = row / 4096
        float    s    = (float)(grow & 4095u);       // position = row % 4096
        float    fl   = (float)lengths[b];
        float    v    = 0.0f;
        if (s < fl) {
            float safe = fmaxf(fl, 1.0f);
            v = cosf((s / safe) * PE_PI_F);          // reference op order
        }
        tile[t * LDS_ROW_F + 512u] = v;              // ds_store_b32
    }

    // Own wave's async loads have landed in LDS; barrier for cross-wave
    // visibility (compiler also waits DScnt for the ds_store at the barrier).
    asm volatile("s_wait_asynccnt 0x0" ::: "memory");
    __syncthreads();

    // ---- Phase 2: async LDS -> global, b32 per lane (rows are 2052 B) -----
#pragma unroll 1
    for (uint32_t r = 0; r < ROWS_PER_BLOCK; ++r) {
        uint32_t lds_row = lds_base + r * (LDS_ROW_F * 4u);
        uint32_t gout    = (row0 + r) * 2052u;       // byte offset into out
#pragma unroll
        for (uint32_t j = 0; j < 3; ++j) {
            uint32_t d = t + j * 256u;               // 0..512 valid
            if (d < 513u) {
                uint32_t lds = lds_row + d * 4u;
                uint32_t gof = gout + d * 4u;
                asm volatile("global_store_async_from_lds_b32 %0, %1, %2 th:TH_STORE_NT"
                             :: "v"(gof), "v"(lds), "s"(out) : "memory");
            }
        }
    }
    // s_endpgm performs an implicit wait-idle, draining outstanding ASYNCcnt.
}

extern "C" void kernel_launch(void* const* d_in, const int* in_sizes, int n_in,
                              void* d_out, int out_size, void* d_ws, size_t ws_size,
                              hipStream_t stream)
{
    (void)in_sizes; (void)n_in; (void)d_ws; (void)ws_size;

    const float* inp     = (const float*)d_in[0];   // fp32 [16,4096,512]
    const int*   lengths = (const int*)d_in[1];     // int32 [16]
    float*       out     = (float*)d_out;           // fp32 [16,4096,513]

    uint32_t rows   = (uint32_t)out_size / 513u;    // 65536
    uint32_t blocks = rows / ROWS_PER_BLOCK;        // 4096

    concat_pe_async_kernel<<<dim3(blocks), dim3(256), 0, stream>>>(inp, lengths, out);
}